// OIoULoss_1760936591611
// MI455X (gfx1250) — compile-verified
//
#include <hip/hip_runtime.h>
#include <hip/hip_bf16.h>
#include <math.h>

#define PI_F 3.1415926f
#define EPS_F 1e-06f

typedef __attribute__((ext_vector_type(2))) float v2f;
typedef __attribute__((ext_vector_type(8))) float v8f;

// ---------------------------------------------------------------------------
// Kernel 1: per-box OIoU loss + block-level deterministic reduction.
// One thread = one box. All 24-vertex state fully unrolled -> registers.
// ---------------------------------------------------------------------------
__global__ void __launch_bounds__(256)
oiou_loss_kernel(const float* __restrict__ pred,
                 const float* __restrict__ target,
                 const float* __restrict__ weight,
                 float* __restrict__ partials,
                 int n) {
    __shared__ float sdata[256];
    const int tid = threadIdx.x;
    const int gid = blockIdx.x * 256 + tid;

    float acc = 0.0f;
    if (gid < n) {
        const float* pr = pred + 5 * gid;
        const float* tr = target + 5 * gid;
        __builtin_prefetch(pr, 0, 0);   // global_prefetch_b8
        __builtin_prefetch(tr, 0, 0);

        const float px = pr[0], py = pr[1], pw = pr[2], ph = pr[3], pa = pr[4];
        const float tx = tr[0], ty = tr[1], tw = tr[2], th = tr[3], ta = tr[4];

        // ---- box2corners ----
        float sp, cp, st, ct;
        sincosf(pa, &sp, &cp);
        sincosf(ta, &st, &ct);
        const float kx[4] = {0.5f, -0.5f, -0.5f, 0.5f};
        const float ky[4] = {0.5f, 0.5f, -0.5f, -0.5f};
        float c1x[4], c1y[4], c2x[4], c2y[4];
#pragma unroll
        for (int k = 0; k < 4; ++k) {
            float ax = kx[k] * pw, ay = ky[k] * ph;
            c1x[k] = ax * cp - ay * sp + px;
            c1y[k] = ax * sp + ay * cp + py;
            float bx = kx[k] * tw, by = ky[k] * th;
            c2x[k] = bx * ct - by * st + tx;
            c2y[k] = bx * st + by * ct + ty;
        }

        // ---- vertex pool: 4 c1 + 4 c2 + 16 edge intersections ----
        float vx[24], vy[24], mk[24];

        // points_in_box(c1, c2): a=c2[0], b=c2[1], d=c2[3]
        {
            float abx = c2x[1] - c2x[0], aby = c2y[1] - c2y[0];
            float adx = c2x[3] - c2x[0], ady = c2y[3] - c2y[0];
            float rab = 1.0f / (abx * abx + aby * aby);
            float rad = 1.0f / (adx * adx + ady * ady);
#pragma unroll
            for (int k = 0; k < 4; ++k) {
                float amx = c1x[k] - c2x[0], amy = c1y[k] - c2y[0];
                float pab = (abx * amx + aby * amy) * rab;
                float pad = (adx * amx + ady * amy) * rad;
                bool m = (pab > -1e-6f) && (pab < 1.0f + 1e-6f) &&
                         (pad > -1e-6f) && (pad < 1.0f + 1e-6f);
                vx[k] = c1x[k]; vy[k] = c1y[k]; mk[k] = m ? 1.0f : 0.0f;
            }
        }
        // points_in_box(c2, c1)
        {
            float abx = c1x[1] - c1x[0], aby = c1y[1] - c1y[0];
            float adx = c1x[3] - c1x[0], ady = c1y[3] - c1y[0];
            float rab = 1.0f / (abx * abx + aby * aby);
            float rad = 1.0f / (adx * adx + ady * ady);
#pragma unroll
            for (int k = 0; k < 4; ++k) {
                float amx = c2x[k] - c1x[0], amy = c2y[k] - c1y[0];
                float pab = (abx * amx + aby * amy) * rab;
                float pad = (adx * amx + ady * amy) * rad;
                bool m = (pab > -1e-6f) && (pab < 1.0f + 1e-6f) &&
                         (pad > -1e-6f) && (pad < 1.0f + 1e-6f);
                vx[4 + k] = c2x[k]; vy[4 + k] = c2y[k]; mk[4 + k] = m ? 1.0f : 0.0f;
            }
        }
        // 16 edge-edge intersections
#pragma unroll
        for (int i = 0; i < 4; ++i) {
            float x1 = c1x[i], y1 = c1y[i];
            float x2 = c1x[(i + 1) & 3], y2 = c1y[(i + 1) & 3];
#pragma unroll
            for (int j = 0; j < 4; ++j) {
                float x3 = c2x[j], y3 = c2y[j];
                float x4 = c2x[(j + 1) & 3], y4 = c2y[(j + 1) & 3];
                float den = (x2 - x1) * (y4 - y3) - (y2 - y1) * (x4 - x3);
                float tn  = (x3 - x1) * (y4 - y3) - (y3 - y1) * (x4 - x3);
                float un  = (x3 - x1) * (y2 - y1) - (y3 - y1) * (x2 - x1);
                float sd  = (den == 0.0f) ? 1.0f : den;
                float r   = 1.0f / sd;
                float t   = tn * r, u = un * r;
                bool m = (den != 0.0f) && (t > 0.0f) && (t < 1.0f) &&
                         (u > 0.0f) && (u < 1.0f);
                float fm = m ? 1.0f : 0.0f;
                int idx = 8 + i * 4 + j;
                vx[idx] = (x1 + t * (x2 - x1)) * fm;
                vy[idx] = (y1 + t * (y2 - y1)) * fm;
                mk[idx] = fm;
            }
        }

        // ---- centroid of valid verts, recentre ----
        float nv = 0.0f, sx = 0.0f, sy = 0.0f;
#pragma unroll
        for (int k = 0; k < 24; ++k) { nv += mk[k]; sx += vx[k] * mk[k]; sy += vy[k] * mk[k]; }
        int   nvi  = (int)nv;
        float rden = 1.0f / fmaxf(nv, 1.0f);
        float mx = sx * rden, my = sy * rden;

        // ---- sort key: diamond pseudo-angle, strictly monotone in atan2 ----
        // Key precision is irrelevant (only the argsort permutation matters),
        // so a raw v_rcp_f32 is fine here. r = x/(|x|+|y|) (r=1 when x=y=0 =>
        // key 0, matching atan2(0,0)=0); key = (y>=0) ? 1-r : r-1 maps
        // (-pi,pi] -> [-2,2] monotonically with the same branch cut
        // (y=0,x<0 -> 2 = max, like atan2 = +pi).
        float ang[24];
#pragma unroll
        for (int k = 0; k < 24; ++k) {
            float ax = (vx[k] - mx) * mk[k];
            float ay = (vy[k] - my) * mk[k];
            vx[k] = ax; vy[k] = ay;
            float s = fabsf(ax) + fabsf(ay);
            float r = (s > 0.0f) ? (ax * __builtin_amdgcn_rcpf(s)) : 1.0f;
            float p = (ay >= 0.0f) ? (1.0f - r) : (r - 1.0f);
            ang[k] = (mk[k] > 0.0f) ? p : 1000000.0f;
        }

        // ---- Batcher odd-even mergesort, n=32 pruned to 24 live slots ----
        // 132 comparators with LITERAL indices (virtual slots 24..31 hold +inf
        // and never exchange, so all comparators with hi>=24 are dropped).
#define CSWAP(LO, HI) do {                                   \
            bool sw_ = ang[LO] > ang[HI];                    \
            float a0_ = sw_ ? ang[HI] : ang[LO];             \
            float a1_ = sw_ ? ang[LO] : ang[HI];             \
            float x0_ = sw_ ? vx[HI] : vx[LO];               \
            float x1_ = sw_ ? vx[LO] : vx[HI];               \
            float y0_ = sw_ ? vy[HI] : vy[LO];               \
            float y1_ = sw_ ? vy[LO] : vy[HI];               \
            ang[LO] = a0_; ang[HI] = a1_;                    \
            vx[LO] = x0_;  vx[HI] = x1_;                     \
            vy[LO] = y0_;  vy[HI] = y1_;                     \
        } while (0)

        // p=1,k=1
        CSWAP(0,1);  CSWAP(2,3);  CSWAP(4,5);  CSWAP(6,7);
        CSWAP(8,9);  CSWAP(10,11);CSWAP(12,13);CSWAP(14,15);
        CSWAP(16,17);CSWAP(18,19);CSWAP(20,21);CSWAP(22,23);
        // p=2,k=2
        CSWAP(0,2);  CSWAP(1,3);  CSWAP(4,6);  CSWAP(5,7);
        CSWAP(8,10); CSWAP(9,11); CSWAP(12,14);CSWAP(13,15);
        CSWAP(16,18);CSWAP(17,19);CSWAP(20,22);CSWAP(21,23);
        // p=2,k=1
        CSWAP(1,2);  CSWAP(5,6);  CSWAP(9,10); CSWAP(13,14);
        CSWAP(17,18);CSWAP(21,22);
        // p=4,k=4
        CSWAP(0,4);  CSWAP(1,5);  CSWAP(2,6);  CSWAP(3,7);
        CSWAP(8,12); CSWAP(9,13); CSWAP(10,14);CSWAP(11,15);
        CSWAP(16,20);CSWAP(17,21);CSWAP(18,22);CSWAP(19,23);
        // p=4,k=2
        CSWAP(2,4);  CSWAP(3,5);  CSWAP(10,12);CSWAP(11,13);
        CSWAP(18,20);CSWAP(19,21);
        // p=4,k=1
        CSWAP(1,2);  CSWAP(3,4);  CSWAP(5,6);  CSWAP(9,10);
        CSWAP(11,12);CSWAP(13,14);CSWAP(17,18);CSWAP(19,20);
        CSWAP(21,22);
        // p=8,k=8
        CSWAP(0,8);  CSWAP(1,9);  CSWAP(2,10); CSWAP(3,11);
        CSWAP(4,12); CSWAP(5,13); CSWAP(6,14); CSWAP(7,15);
        // p=8,k=4
        CSWAP(4,8);  CSWAP(5,9);  CSWAP(6,10); CSWAP(7,11);
        // p=8,k=2
        CSWAP(2,4);  CSWAP(3,5);  CSWAP(6,8);  CSWAP(7,9);
        CSWAP(10,12);CSWAP(11,13);CSWAP(18,20);CSWAP(19,21);
        // p=8,k=1
        CSWAP(1,2);  CSWAP(3,4);  CSWAP(5,6);  CSWAP(7,8);
        CSWAP(9,10); CSWAP(11,12);CSWAP(13,14);CSWAP(17,18);
        CSWAP(19,20);CSWAP(21,22);
        // p=16,k=16
        CSWAP(0,16); CSWAP(1,17); CSWAP(2,18); CSWAP(3,19);
        CSWAP(4,20); CSWAP(5,21); CSWAP(6,22); CSWAP(7,23);
        // p=16,k=8
        CSWAP(8,16); CSWAP(9,17); CSWAP(10,18);CSWAP(11,19);
        CSWAP(12,20);CSWAP(13,21);CSWAP(14,22);CSWAP(15,23);
        // p=16,k=4
        CSWAP(4,8);  CSWAP(5,9);  CSWAP(6,10); CSWAP(7,11);
        CSWAP(12,16);CSWAP(13,17);CSWAP(14,18);CSWAP(15,19);
        // p=16,k=2
        CSWAP(2,4);  CSWAP(3,5);  CSWAP(6,8);  CSWAP(7,9);
        CSWAP(10,12);CSWAP(11,13);CSWAP(14,16);CSWAP(15,17);
        CSWAP(18,20);CSWAP(19,21);
        // p=16,k=1
        CSWAP(1,2);  CSWAP(3,4);  CSWAP(5,6);  CSWAP(7,8);
        CSWAP(9,10); CSWAP(11,12);CSWAP(13,14);CSWAP(15,16);
        CSWAP(17,18);CSWAP(19,20);CSWAP(21,22);
#undef CSWAP

        // ---- shoelace over sorted chain + closing edge ----
        float area = 0.0f;
#pragma unroll
        for (int k = 0; k < 23; ++k)
            area += vx[k] * vy[k + 1] - vx[k + 1] * vy[k];
        int last = nvi - 1; if (last < 0) last = 0;
        float lx = 0.0f, ly = 0.0f;
#pragma unroll
        for (int k = 0; k < 24; ++k)
            if (k == last) { lx = vx[k]; ly = vy[k]; }
        area += lx * vy[0] - vx[0] * ly;
        float inter = fabsf(area) * 0.5f;

        // ---- IoU ----
        float a1 = pw * ph, a2 = tw * th;
        float iou = inter / (a1 + a2 - inter);
        float ious = fmaxf(iou, EPS_F);

        // ---- distance term ----
        float dth = (pa - ta) * (1.0f / PI_F);
        float cpa = fabsf(cp), spa = fabsf(sp);
        float cta = fabsf(ct), sta = fabsf(st);
        float hwp = (pw * cpa + ph * spa) * 0.5f, hhp = (pw * spa + ph * cpa) * 0.5f;
        float hwt = (tw * cta + th * sta) * 0.5f, hht = (tw * sta + th * cta) * 0.5f;
        float b1x1 = px - hwp, b1y1 = py - hhp, b1x2 = px + hwp, b1y2 = py + hhp;
        float b2x1 = tx - hwt, b2y1 = ty - hht, b2x2 = tx + hwt, b2y2 = ty + hht;
        float ddx = (b2x1 + b2x2 - (b1x1 + b1x2)); ddx = ddx * ddx * 0.25f;
        float ddy = (b2y1 + b2y2 - (b1y1 + b1y2)); ddy = ddy * ddy * 0.25f;
        float ex1 = fminf(b1x1, b2x1), ey1 = fminf(b1y1, b2y1);
        float ex2 = fmaxf(b1x2, b2x2), ey2 = fmaxf(b1y2, b2y2);
        float ew = fmaxf(ex2 - ex1, 0.0f), eh = fmaxf(ey2 - ey1, 0.0f);
        float cc = ew * ew + eh * eh + EPS_F;
        float dist = (ddx + ddy) / cc * expf(fabsf(dth) - 1.0f);

        float loss = 1.0f - ious + dist;
        acc = loss * weight[gid];
    }

    // ---- deterministic block reduction (fixed pairing tree) ----
    sdata[tid] = acc;
    __syncthreads();
#pragma unroll
    for (int s = 128; s > 0; s >>= 1) {
        if (tid < s) sdata[tid] += sdata[tid + s];
        __syncthreads();
    }
    if (tid == 0) partials[blockIdx.x] = sdata[0];
}

// ---------------------------------------------------------------------------
// Kernel 2: single wave32. Each lane sums partials in fixed stride order, then
// the 32 lane sums are reduced with V_WMMA_F32_16X16X4_F32 (A = ones 16x4,
// B = lane sums). Since A is all ones, D[0][n] = column-sum(n) regardless of
// the exact KxN lane mapping of B, and sum over D row 0 (lanes 0..15 of c[0])
// is the exact fp32 total. Deterministic fixed-order readout via __shfl.
// ---------------------------------------------------------------------------
__global__ void __launch_bounds__(32)
oiou_reduce_wmma(const float* __restrict__ partials, int nPartials,
                 float* __restrict__ out, float invN) {
    const int lane = threadIdx.x;
    float s = 0.0f;
    for (int i = lane; i < nPartials; i += 32) s += partials[i];

    v2f a; a.x = 1.0f; a.y = 1.0f;   // A (16x4 f32): all ones
    v2f b; b.x = s;    b.y = 0.0f;   // B (4x16 f32): 32 lane sums + zeros
    v8f c = {};
    c = __builtin_amdgcn_wmma_f32_16x16x4_f32(
            /*neg_a=*/false, a, /*neg_b=*/false, b,
            /*c_mod=*/(short)0, c, /*reuse_a=*/false, /*reuse_b=*/false);

    float total = 0.0f;
#pragma unroll
    for (int i = 0; i < 16; ++i)
        total += __shfl(c[0], i, 32);   // D[0][i], fixed order

    if (lane == 0) out[0] = total * invN;
}

// ---------------------------------------------------------------------------
extern "C" void kernel_launch(void* const* d_in, const int* in_sizes, int n_in,
                              void* d_out, int out_size, void* d_ws, size_t ws_size,
                              hipStream_t stream) {
    const float* pred   = (const float*)d_in[0];
    const float* target = (const float*)d_in[1];
    const float* weight = (const float*)d_in[2];
    const int n = in_sizes[2];          // weight holds n elements

    const int threads = 256;
    const int blocks  = (n + threads - 1) / threads;
    float* partials = (float*)d_ws;     // blocks * 4 bytes scratch

    oiou_loss_kernel<<<blocks, threads, 0, stream>>>(pred, target, weight, partials, n);
    oiou_reduce_wmma<<<1, 32, 0, stream>>>(partials, blocks, (float*)d_out, 1.0f / (float)n);
}